// ActivationGraphSageEdgeReprLayer_50027779064262
// MI455X (gfx1250) — compile-verified
//
#include <hip/hip_runtime.h>
#include <cstdint>

// GatedGCN-style edge-repr layer: N nodes, DEG=16 in-edges/node (dst sorted,
// exactly DEG consecutive edges per node), D=64 features.
// Pure bandwidth kernel (~930 MB HBM traffic, ~1.3 GFLOP) -> optimize data
// movement: async global->LDS edge-tile staging, NT stores, wave32/node.

#define DEG 16
#define DD  64
#define NPB 8              // nodes (waves) per 256-thread block
#define L2EPS 1e-12f

typedef float v2f __attribute__((ext_vector_type(2)));
typedef int   v4i __attribute__((ext_vector_type(4)));
typedef __attribute__((address_space(1))) v4i* g_v4i_ptr;
typedef __attribute__((address_space(3))) v4i* l_v4i_ptr;

__device__ __forceinline__ void async_copy_b128(const float* gptr, float* lptr) {
#if __has_builtin(__builtin_amdgcn_global_load_async_to_lds_b128)
  // Builtin signature: (int4 AS(1)*, int4 AS(3)*, imm offset, imm cpol).
  // AS(3) ptrs are 32-bit; low 32 bits of a flat shared ptr = LDS byte offset.
  g_v4i_ptr gp = (g_v4i_ptr)(uintptr_t)gptr;
  unsigned  lo = (unsigned)(uintptr_t)lptr;
  l_v4i_ptr lp = (l_v4i_ptr)lo;
  __builtin_amdgcn_global_load_async_to_lds_b128(gp, lp, /*offset=*/0, /*cpol=*/0);
#else
  // ISA 15.18.3 opcode 98: VDST = LDS byte address VGPR, VADDR = 64-bit global
  // address VGPR pair, GV mode (no SADDR).
  unsigned lds_off = (unsigned)(uintptr_t)lptr;
  unsigned long long ga = (unsigned long long)(uintptr_t)gptr;
  asm volatile("global_load_async_to_lds_b128 %0, %1, off"
               :: "v"(lds_off), "v"(ga) : "memory");
#endif
}

__device__ __forceinline__ void wait_async0() {
#if __has_builtin(__builtin_amdgcn_s_wait_asynccnt)
  __builtin_amdgcn_s_wait_asynccnt(0);
#else
  asm volatile("s_wait_asynccnt 0x0" ::: "memory");
#endif
}

__global__ __launch_bounds__(NPB * 32, 2)
void gatedgcn_node_kernel(const float* __restrict__ h,
                          const float* __restrict__ e,
                          const float* __restrict__ norm,
                          const int*   __restrict__ src,
                          float* __restrict__ h_out,
                          float* __restrict__ b_out,
                          float* __restrict__ e_out,
                          int n) {
  __shared__ __align__(16) float ldse[NPB * DEG * DD];   // 8 x 4KB edge tiles
  const int wave = threadIdx.x >> 5;
  const int lane = threadIdx.x & 31;
  const int node = blockIdx.x * NPB + wave;              // wave-uniform
  if (node >= n) return;

  // ---- stage this node's 16x64 f32 edge tile (4 KB contiguous) into LDS ----
  float* etile = &ldse[wave * (DEG * DD)];
  const float* eg = e + (size_t)node * (DEG * DD);
#pragma unroll
  for (int i = 0; i < 8; ++i)                            // 8 x (32 lanes x 16B)
    async_copy_b128(eg + i * 128 + lane * 4, etile + i * 128 + lane * 4);

  // ---- overlap: load own features while the DMA streams ----
  const float nrm = norm[node];                          // scalar (uniform)
  const v2f* h2 = (const v2f*)h;                         // lane t -> feats 2t,2t+1
  v2f hnd = h2[(size_t)node * 32 + lane] * nrm;

  v2f c = {0.f, 0.f};                                    // msg >= 0 => max init 0
  v2f* eout2 = (v2f*)e_out;
  const v2f* et2 = (const v2f*)etile;

  wait_async0();

#pragma unroll 4
  for (int k = 0; k < DEG; ++k) {
    const int   s  = src[node * DEG + k];                // lane-uniform -> s_load
    const float ns = norm[s];                            // lane-uniform -> s_load
    v2f hns = h2[(size_t)s * 32 + lane] * ns;            // L2-resident gather
    v2f eij = et2[k * 32 + lane] + hns + hnd;            // ds_load_b64
    v2f eo  = {fmaxf(eij.x, 0.f), fmaxf(eij.y, 0.f)};
    __builtin_nontemporal_store(eo, &eout2[((size_t)node * DEG + k) * 32 + lane]);
    v2f sg  = {1.f / (1.f + __expf(-eij.x)),
               1.f / (1.f + __expf(-eij.y))};
    v2f m   = sg * hns;
    c.x = fmaxf(c.x, fmaxf(m.x, 0.f));
    c.y = fmaxf(c.y, fmaxf(m.y, 0.f));
  }

  // ---- L2 norm of bundle = [hnd(64) | c(64)] : wave32 shuffle reduction ----
  float ss = hnd.x * hnd.x + hnd.y * hnd.y + c.x * c.x + c.y * c.y;
#pragma unroll
  for (int off = 16; off > 0; off >>= 1)
    ss += __shfl_xor(ss, off, 32);
  const float inv = 1.f / fmaxf(sqrtf(ss), L2EPS);

  // ---- outputs ----
  ((v2f*)h_out)[(size_t)node * 32 + lane] = c * nrm;

  v2f b0 = hnd * inv;
  v2f b1 = c * inv;
  b0.x = fmaxf(b0.x, 0.f); b0.y = fmaxf(b0.y, 0.f);
  b1.x = fmaxf(b1.x, 0.f); b1.y = fmaxf(b1.y, 0.f);
  v2f* b2 = (v2f*)b_out;                                 // row = 128 floats
  b2[(size_t)node * 64 + lane]      = b0;
  b2[(size_t)node * 64 + 32 + lane] = b1;
}

extern "C" void kernel_launch(void* const* d_in, const int* in_sizes, int n_in,
                              void* d_out, int out_size, void* d_ws, size_t ws_size,
                              hipStream_t stream) {
  const float* h    = (const float*)d_in[0];   // [N, 64]
  const float* e    = (const float*)d_in[1];   // [N*16, 64]
  const float* norm = (const float*)d_in[2];   // [N, 1]
  const int*   src  = (const int*)  d_in[3];   // [N*16]
  // d_in[4] = dst = repeat(arange(N), 16): implicit in the layout.
  const int n = in_sizes[2];                   // N (norm has N elements)

  float* out   = (float*)d_out;                // (h_out, b, e_out) concatenated
  float* h_out = out;                          // N*64
  float* b_out = out + (size_t)n * DD;         // N*128
  float* e_out = out + (size_t)n * DD * 3;     // N*16*64

  const int blocks = (n + NPB - 1) / NPB;      // 12500 for N=100000
  gatedgcn_node_kernel<<<blocks, NPB * 32, 0, stream>>>(
      h, e, norm, src, h_out, b_out, e_out, n);
}